// MeshCodec_89206470738132
// MI455X (gfx1250) — compile-verified
//
#include <hip/hip_runtime.h>
#include <math.h>

typedef __attribute__((ext_vector_type(2))) float v2f;
typedef __attribute__((ext_vector_type(8))) float v8f;

#define NCOLS 1056   // 576 coor + 48 angle + 192 normal + 16 area + 192 emangle + 16 emnoangle + 16 emfreq
#define KDIM  20     // 9 coords + 3 angles + 3 normal + 1 area + 1 emnoangle + 3 zero pad
#define DEG2RAD 0.017453292519943295f
#define EPSC 1e-5f

// ---------------------------------------------------------------------------
// Kernel 1: build block-diagonal weight matrix Wbig (KDIM x NCOLS, row major)
// and per-batch bias vector (NCOLS) with the batch-constant embeddings folded in.
// ---------------------------------------------------------------------------
__global__ void mesh_prep(const float* __restrict__ W_angle, const float* __restrict__ b_angle,
                          const float* __restrict__ W_area,  const float* __restrict__ b_area,
                          const float* __restrict__ W_normal,const float* __restrict__ b_normal,
                          const float* __restrict__ W_emnoangle, const float* __restrict__ b_emnoangle,
                          const float* __restrict__ W_emangle,   const float* __restrict__ b_emangle,
                          const float* __restrict__ W_emfreq,    const float* __restrict__ b_emfreq,
                          const float* __restrict__ W_coor,  const float* __restrict__ b_coor,
                          const float* __restrict__ theta, const float* __restrict__ phi,
                          const float* __restrict__ freq,
                          float* __restrict__ Wbig, float* __restrict__ bias, int Bn) {
    int tid = blockIdx.x * blockDim.x + threadIdx.x;
    const int NW = KDIM * NCOLS;
    if (tid < NW) {
        int k = tid / NCOLS, n = tid % NCOLS;
        float w = 0.0f;
        if (n < 576)        { if (k < 9)             w = W_coor[k * 576 + n]; }
        else if (n < 624)   { if (k >= 9 && k < 12)  w = W_angle[(k - 9) * 48 + (n - 576)]; }
        else if (n < 816)   { if (k >= 12 && k < 15) w = W_normal[(k - 12) * 192 + (n - 624)]; }
        else if (n < 832)   { if (k == 15)           w = W_area[n - 816]; }
        // 832..1023: emangle (batch-constant, in bias)  -> 0
        else if (n >= 1024 && n < 1040) { if (k == 16) w = W_emnoangle[n - 1024]; }
        // 1040..1055: emfreq (batch-constant, in bias)  -> 0
        Wbig[tid] = w;
    } else {
        int t = tid - NW;
        if (t >= Bn * NCOLS) return;
        int b = t / NCOLS, n = t % NCOLS;
        float val;
        if (n < 576)      val = b_coor[n];
        else if (n < 624) val = b_angle[n - 576];
        else if (n < 816) val = b_normal[n - 624];
        else if (n < 832) val = b_area[n - 816];
        else if (n < 1024) {
            int j = n - 832;
            float th = theta[b] * DEG2RAD;
            float ph = phi[b]   * DEG2RAD;
            float ix = sinf(ph) * cosf(th);
            float iy = sinf(ph) * sinf(th);
            float iz = cosf(ph);
            val = ix * W_emangle[j] + iy * W_emangle[192 + j] + iz * W_emangle[384 + j] + b_emangle[j];
        } else if (n < 1040) {
            val = b_emnoangle[n - 1024];
        } else {
            int j = n - 1040;
            float fl = (log10f(freq[b]) + 1.0f) * 0.5f;   // (log10(f)-log10(0.1))/(log10(10)-log10(0.1))
            val = fl * W_emfreq[j] + b_emfreq[j];
        }
        bias[b * NCOLS + n] = val;
    }
}

// ---------------------------------------------------------------------------
// Kernel 2: per-face geometry -> 20-float feature rows
//   [v0.xyz v1.xyz v2.xyz  ang0 ang1 ang2  nx ny nz  area  emnoangle  0 0 0]
// ---------------------------------------------------------------------------
__global__ void mesh_geom(const float* __restrict__ vertices, const int* __restrict__ faces,
                          const float* __restrict__ theta, const float* __restrict__ phi,
                          float* __restrict__ Feat, int Bn, int V, int F) {
    int gid = blockIdx.x * blockDim.x + threadIdx.x;
    if (gid >= Bn * F) return;
    int b = gid / F;
    int f = gid - b * F;

    const int* fp = faces + ((size_t)b * F + f) * 3;
    int i0 = fp[0], i1 = fp[1], i2 = fp[2];
    const float* vb = vertices + (size_t)b * V * 3;
    float v0x = vb[i0 * 3], v0y = vb[i0 * 3 + 1], v0z = vb[i0 * 3 + 2];
    float v1x = vb[i1 * 3], v1y = vb[i1 * 3 + 1], v1z = vb[i1 * 3 + 2];
    float v2x = vb[i2 * 3], v2y = vb[i2 * 3 + 1], v2z = vb[i2 * 3 + 2];

    // edges: e0=v0-v2, e1=v1-v0, e2=v2-v1
    float e0x = v0x - v2x, e0y = v0y - v2y, e0z = v0z - v2z;
    float e1x = v1x - v0x, e1y = v1y - v0y, e1z = v1z - v0z;
    float e2x = v2x - v1x, e2y = v2y - v1y, e2z = v2z - v1z;

    float n0 = sqrtf(e0x * e0x + e0y * e0y + e0z * e0z);
    float n1 = sqrtf(e1x * e1x + e1y * e1y + e1z * e1z);
    float n2 = sqrtf(e2x * e2x + e2y * e2y + e2z * e2z);
    float r0 = 1.0f / fmaxf(n0, 1e-12f);
    float r1 = 1.0f / fmaxf(n1, 1e-12f);
    float r2 = 1.0f / fmaxf(n2, 1e-12f);
    float a0x = e0x * r0, a0y = e0y * r0, a0z = e0z * r0;
    float a1x = e1x * r1, a1y = e1y * r1, a1z = e1z * r1;
    float a2x = e2x * r2, a2y = e2y * r2, a2z = e2z * r2;

    // reference rolls the COORDINATE axis: nd[c] = -sum_i nv[i][c]*nv[i][(c+2)%3]
    float nd0 = -(a0x * a0z + a1x * a1z + a2x * a2z);
    float nd1 = -(a0y * a0x + a1y * a1x + a2y * a2x);
    float nd2 = -(a0z * a0y + a1z * a1y + a2z * a2y);
    float ang0 = acosf(fminf(fmaxf(nd0, -1.0f + EPSC), 1.0f - EPSC));
    float ang1 = acosf(fminf(fmaxf(nd1, -1.0f + EPSC), 1.0f - EPSC));
    float ang2 = acosf(fminf(fmaxf(nd2, -1.0f + EPSC), 1.0f - EPSC));

    // cross(e0, e1)
    float cx = e0y * e1z - e0z * e1y;
    float cy = e0z * e1x - e0x * e1z;
    float cz = e0x * e1y - e0y * e1x;
    float clen = sqrtf(cx * cx + cy * cy + cz * cz);
    float area = 0.5f * clen;
    float rinv = 1.0f / fmaxf(clen, 1e-12f);
    float nx = cx * rinv, ny = cy * rinv, nz = cz * rinv;

    float th = theta[b] * DEG2RAD;
    float ph = phi[b]   * DEG2RAD;
    float ix = sinf(ph) * cosf(th);
    float iy = sinf(ph) * sinf(th);
    float iz = cosf(ph);                       // already unit length
    float d2 = -(nx * ix + ny * iy + nz * iz);
    float emno = acosf(fminf(fmaxf(d2, -1.0f + EPSC), 1.0f - EPSC));

    float4* row = (float4*)(Feat + (size_t)gid * KDIM);
    row[0] = make_float4(v0x, v0y, v0z, v1x);
    row[1] = make_float4(v1y, v1z, v2x, v2y);
    row[2] = make_float4(v2z, ang0, ang1, ang2);
    row[3] = make_float4(nx, ny, nz, area);
    row[4] = make_float4(emno, 0.0f, 0.0f, 0.0f);
}

// ---------------------------------------------------------------------------
// Kernel 3: GEMM  out[b] = Feat[b] (F x 20) @ Wbig (20 x 1056) + bias[b]
// one wave per 16x16 tile, K in 5 steps of V_WMMA_F32_16X16X4_F32
// ---------------------------------------------------------------------------
__global__ void mesh_gemm(const float* __restrict__ Feat, const float* __restrict__ Wbig,
                          const float* __restrict__ bias, float* __restrict__ out, int F) {
    const int lane  = threadIdx.x & 31;
    const int wave  = threadIdx.x >> 5;
    const int ntile = blockIdx.x * 8 + wave;
    if (ntile >= NCOLS / 16) return;           // wave-uniform: EXEC stays all-ones
    const int mtile  = blockIdx.y;
    const int b      = blockIdx.z;
    const int n_base = ntile * 16;
    const int m_base = mtile * 16;
    const int lo = lane & 15;
    const int hi = lane >> 4;                  // K half-select (0 / 1)

    int m = m_base + lo;
    if (m > F - 1) m = F - 1;                  // clamp A loads (F%16==0 normally)
    const float* featRow = Feat + ((size_t)b * F + m) * KDIM + 2 * hi;

    // bias depends only on column -> preload accumulator with it
    float bv = bias[b * NCOLS + n_base + lo];
    v8f c = {bv, bv, bv, bv, bv, bv, bv, bv};

    const float* wcol = Wbig + 2 * hi * NCOLS + n_base + lo;
#pragma unroll
    for (int s = 0; s < 5; ++s) {
        // A fragment (16x4 f32): lanes 0-15 = K(4s),K(4s+1); lanes 16-31 = K(4s+2),K(4s+3)
        v2f a = *(const v2f*)(featRow + 4 * s);
        // B fragment (4x16 f32): VGPR0 = rows 4s / 4s+2, VGPR1 = rows 4s+1 / 4s+3
        v2f bm;
        bm.x = wcol[(4 * s) * NCOLS];
        bm.y = wcol[(4 * s + 1) * NCOLS];
        c = __builtin_amdgcn_wmma_f32_16x16x4_f32(false, a, false, bm,
                                                  (short)0, c, false, false);
    }

    // C/D layout: VGPR r -> (M=r, N=lane) for lanes 0-15, (M=r+8, N=lane-16) for 16-31
    float* orow = out + ((size_t)b * F + m_base) * NCOLS + n_base + lo;
    if (m_base + 16 <= F) {
        // full tile (always true when F % 16 == 0): straight-line streaming stores,
        // non-temporal so the 676 MB output does not rinse L2 (which holds Wbig/Feat)
#pragma unroll
        for (int r = 0; r < 8; ++r) {
            __builtin_nontemporal_store(c[r], orow + (size_t)(r + 8 * hi) * NCOLS);
        }
    } else {
        // ragged last M-tile
#pragma unroll
        for (int r = 0; r < 8; ++r) {
            int mr = r + 8 * hi;
            if (m_base + mr < F)
                orow[(size_t)mr * NCOLS] = c[r];
        }
    }
}

// ---------------------------------------------------------------------------
extern "C" void kernel_launch(void* const* d_in, const int* in_sizes, int n_in,
                              void* d_out, int out_size, void* d_ws, size_t ws_size,
                              hipStream_t stream) {
    const float* vertices    = (const float*)d_in[0];
    const int*   faces       = (const int*)  d_in[1];
    const float* theta       = (const float*)d_in[2];
    const float* phi         = (const float*)d_in[3];
    const float* freq        = (const float*)d_in[4];
    const float* W_angle     = (const float*)d_in[5];
    const float* b_angle     = (const float*)d_in[6];
    const float* W_area      = (const float*)d_in[7];
    const float* b_area      = (const float*)d_in[8];
    const float* W_normal    = (const float*)d_in[9];
    const float* b_normal    = (const float*)d_in[10];
    const float* W_emnoangle = (const float*)d_in[11];
    const float* b_emnoangle = (const float*)d_in[12];
    const float* W_emangle   = (const float*)d_in[13];
    const float* b_emangle   = (const float*)d_in[14];
    const float* W_emfreq    = (const float*)d_in[15];
    const float* b_emfreq    = (const float*)d_in[16];
    const float* W_coor      = (const float*)d_in[17];
    const float* b_coor      = (const float*)d_in[18];
    float* out = (float*)d_out;

    const int B = in_sizes[2];                 // theta has B elements
    const int V = in_sizes[0] / (3 * B);
    const int F = in_sizes[1] / (3 * B);

    float* ws   = (float*)d_ws;
    float* Wbig = ws;                          // KDIM * NCOLS        = 21120 floats
    float* bias = ws + KDIM * NCOLS;           // B * NCOLS           =  8448 floats
    float* Feat = ws + KDIM * NCOLS + B * NCOLS; // B * F * KDIM (16B-aligned: 29568*4 % 16 == 0)

    // 1) weights + per-batch bias
    {
        int total = KDIM * NCOLS + B * NCOLS;
        mesh_prep<<<(total + 255) / 256, 256, 0, stream>>>(
            W_angle, b_angle, W_area, b_area, W_normal, b_normal,
            W_emnoangle, b_emnoangle, W_emangle, b_emangle,
            W_emfreq, b_emfreq, W_coor, b_coor,
            theta, phi, freq, Wbig, bias, B);
    }
    // 2) per-face features
    {
        int total = B * F;
        mesh_geom<<<(total + 255) / 256, 256, 0, stream>>>(
            vertices, faces, theta, phi, Feat, B, V, F);
    }
    // 3) WMMA GEMM + bias -> output
    {
        int mtiles = (F + 15) / 16;
        dim3 grid((NCOLS / 16 + 7) / 8, mtiles, B);   // 9 x 1250 x 8
        mesh_gemm<<<grid, 256, 0, stream>>>(Feat, Wbig, bias, out, F);
    }
}